// FlowCalculator_64072322121829
// MI455X (gfx1250) — compile-verified
//
#include <hip/hip_runtime.h>
#include <cstdint>

// Problem constants (from reference)
#define B_    32
#define H_    512
#define W_    384
#define V_    6890
#define F_    13776
#define PAD_  64
#define WP_   512                    // W + 2*pad
#define WPB_  ((F_ + 31) / 32)       // 431 u32 words per batch bitmap
#define FACE3_ (F_ * 3)              // 41328 ints = 165,312 bytes
#define NPIX_ (B_ * H_ * WP_)        // 8,388,608 output pixels
#define GRID_ELEMS_ ((size_t)NPIX_ * 2)

// ---- gfx1250 async global->LDS staging (guarded; falls back to int4 copy) ----
#if defined(__gfx1250__) && __has_builtin(__builtin_amdgcn_global_load_async_to_lds_b128) && __has_builtin(__builtin_amdgcn_s_wait_asynccnt)
#define USE_ASYNC_LDS 1
typedef int v4i_ __attribute__((vector_size(4 * sizeof(int))));
typedef __attribute__((address_space(1))) v4i_* g_v4i_p;   // pointer to AS1 int4
typedef __attribute__((address_space(3))) v4i_* l_v4i_p;   // pointer to AS3 int4
#else
#define USE_ASYNC_LDS 0
#endif

// ------------------- kernel 1: zero the visibility bitmap -------------------
__global__ void k_zero_bitmap(uint32_t* __restrict__ bm, int n) {
    int i = blockIdx.x * blockDim.x + threadIdx.x;
    if (i < n) bm[i] = 0u;
}

// ------------- kernel 2: build per-batch face-visibility bitmap -------------
// One block handles 1/32 of one batch's image; dedupe bits in LDS (1.7KB),
// then merge to global with at most 431 atomics per block.
__global__ void k_visibility(const int* __restrict__ vis2, uint32_t* __restrict__ gbm) {
    __shared__ uint32_t sbm[WPB_];
    const int CH  = 32;
    const int PPC = (H_ * W_) / CH;      // 6144 pixels per chunk
    const int b = blockIdx.x / CH;
    const int c = blockIdx.x % CH;

    for (int i = threadIdx.x; i < WPB_; i += blockDim.x) sbm[i] = 0u;
    __syncthreads();

    const int base = b * (H_ * W_) + c * PPC;
    for (int t = threadIdx.x; t < PPC; t += blockDim.x) {
        int v = vis2[base + t];
        if (v >= 0) {
            v = (v < F_) ? v : (F_ - 1);          // matches jnp.clip(flat,0,F-1)
            atomicOr(&sbm[v >> 5], 1u << (v & 31));
        }
    }
    __syncthreads();

    uint32_t* g = gbm + b * WPB_;
    for (int i = threadIdx.x; i < WPB_; i += blockDim.x) {
        uint32_t wv = sbm[i];
        if (wv) atomicOr(&g[i], wv);
    }
}

// -------- kernel 3: fused correspondence + flow-grid + mask, one write ------
__global__ __launch_bounds__(1024)
void k_main(const float* __restrict__ verts1, const int* __restrict__ vis1,
            const float* __restrict__ verts2, const int* __restrict__ face,
            const uint32_t* __restrict__ gbm, float* __restrict__ out) {
    // Stage the whole face table (165KB) into LDS: converts the middle hop of
    // the dependent gather chain vis -> face -> verts into an LDS access.
    __shared__ __align__(16) int sface[FACE3_];

#if USE_ASYNC_LDS
    {
        const int NCH = FACE3_ / 4;  // 10332 x 16B chunks
        for (int i = threadIdx.x; i < NCH; i += blockDim.x) {
            __builtin_amdgcn_global_load_async_to_lds_b128(
                (g_v4i_p)(unsigned long long)(face + i * 4),
                (l_v4i_p)(unsigned int)(unsigned long long)(&sface[i * 4]),
                0, 0);
        }
        __builtin_amdgcn_s_wait_asynccnt(0);
    }
#else
    {
        const int4* f4 = (const int4*)face;
        int4*       s4 = (int4*)sface;
        for (int i = threadIdx.x; i < FACE3_ / 4; i += blockDim.x) s4[i] = f4[i];
    }
#endif
    __syncthreads();

    float* __restrict__ out_mask = out + GRID_ELEMS_;
    const int stride = gridDim.x * blockDim.x;

    for (int idx = blockIdx.x * blockDim.x + threadIdx.x; idx < NPIX_; idx += stride) {
#pragma clang fp contract(off)   // match XLA's separate mul/add rounding
        const int w = idx & (WP_ - 1);
        const int h = (idx >> 9) & (H_ - 1);
        const int b = idx >> 18;

        float gx, gy, m;
        const int wi = w - PAD_;
        if ((unsigned)wi >= (unsigned)W_) {
            // zero-padded side columns: corr=(0,0) -> grid=(-1,-1), mask=2
            gx = -1.0f; gy = -1.0f; m = 2.0f;
        } else {
            const int f = vis1[(b * H_ + h) * W_ + wi];
            if (f < 0) {
                // invalid pixel: corr=(pad,0) -> gx = 2*pad/511-1, gy=-1, mask=2
                gx = (2.0f * (float)PAD_) / 511.0f - 1.0f;
                gy = -1.0f;
                m  = 2.0f;
            } else {
                const int fc = (f < F_) ? f : (F_ - 1);
                const int i0 = sface[3 * fc + 0];
                const int i1 = sface[3 * fc + 1];
                const int i2 = sface[3 * fc + 2];

                const size_t vb = (size_t)b * V_;
                const float* va = verts1 + (vb + i0) * 3;
                const float* vB = verts1 + (vb + i1) * 3;
                const float* vc = verts1 + (vb + i2) * 3;
                const float ax = va[0], ay = va[1];
                const float bx = vB[0], by = vB[1];
                const float cx = vc[0], cy = vc[1];

                float denom = (by - cy) * (ax - cx) + (cx - bx) * (ay - cy);
                if (fabsf(denom) < 1e-8f) denom = 1e-8f;

                const float px = (float)wi, py = (float)h;
                const float w0 = ((by - cy) * (px - cx) + (cx - bx) * (py - cy)) / denom;
                const float w1 = ((cy - ay) * (px - cx) + (ax - cx) * (py - cy)) / denom;
                const float w2 = 1.0f - w0 - w1;

                const float* ua = verts2 + (vb + i0) * 3;
                const float* ub = verts2 + (vb + i1) * 3;
                const float* uc = verts2 + (vb + i2) * 3;
                const float ux = w0 * ua[0] + w1 * ub[0] + w2 * uc[0];
                const float uy = w0 * ua[1] + w1 * ub[1] + w2 * uc[1];

                // Reference: corr = uv + [pad,0]; flow = corr - coord;
                //            g = 2*(coord + flow)/511 - 1   (round-trip kept!)
                const float corrx = ux + (float)PAD_;
                const float wf = (float)w, hf = (float)h;
                const float fu = corrx - wf;
                const float fv = uy - hf;
                gx = (2.0f * (wf + fu)) / 511.0f - 1.0f;
                gy = (2.0f * (hf + fv)) / 511.0f - 1.0f;

                const uint32_t bit = (gbm[b * WPB_ + (fc >> 5)] >> (fc & 31)) & 1u;
                m = bit ? 0.0f : 1.0f;
            }
        }

        ((float2*)out)[idx] = make_float2(gx, gy);  // grid (…,2), coalesced b64 stores
        out_mask[idx] = m;                          // mask (…,1)
    }
}

// --------------------------------- launcher ---------------------------------
extern "C" void kernel_launch(void* const* d_in, const int* in_sizes, int n_in,
                              void* d_out, int out_size, void* d_ws, size_t ws_size,
                              hipStream_t stream) {
    const float* verts1 = (const float*)d_in[0];
    const int*   vis1   = (const int*)  d_in[1];
    const float* verts2 = (const float*)d_in[2];
    const int*   vis2   = (const int*)  d_in[3];
    const int*   face   = (const int*)  d_in[4];
    float*       out    = (float*)d_out;
    uint32_t*    bm     = (uint32_t*)d_ws;          // 32*431 u32 = 55,168 B

    const int nbm = B_ * WPB_;
    k_zero_bitmap<<<(nbm + 255) / 256, 256, 0, stream>>>(bm, nbm);
    k_visibility<<<B_ * 32, 256, 0, stream>>>(vis2, bm);
    k_main<<<512, 1024, 0, stream>>>(verts1, vis1, verts2, face, bm, out);
}